// SAGE_4904852652850
// MI455X (gfx1250) — compile-verified
//
#include <hip/hip_runtime.h>
#include <hip/hip_bf16.h>
#include <math.h>

typedef float v2f __attribute__((ext_vector_type(2)));
typedef float v8f __attribute__((ext_vector_type(8)));

#define N_NODES 50000
#define N_EDGES 800000
#define N_GRAPHS 500
#define IN_CH 128
#define HID 256
#define OUT_CH 5

// full-precision fp32 WMMA: D = A(16x4) * B(4x16) + C(16x16)
#define WMMA_F32(A, B, C) \
  __builtin_amdgcn_wmma_f32_16x16x4_f32(false, (A), false, (B), (short)0, (C), false, false)

// ---------------------------------------------------------------- utilities
__global__ void zero_f(float* __restrict__ p, size_t n) {
  size_t i = (size_t)blockIdx.x * blockDim.x + threadIdx.x;
  if (i < n) p[i] = 0.0f;
}

__global__ void compute_deg(const int* __restrict__ dst, float* __restrict__ deg, int E) {
  int e = blockIdx.x * blockDim.x + threadIdx.x;
  if (e < E) atomicAdd(&deg[dst[e]], 1.0f);
}

// agg[dst] += h[src]; one block per edge, thread per feature
__global__ void scatter_add(const float* __restrict__ h, const int* __restrict__ src,
                            const int* __restrict__ dst, float* __restrict__ agg, int D) {
  int e = blockIdx.x;
  int s = src[e];
  int d = dst[e];
  const float* __restrict__ hs = h + (size_t)s * D;
  float* __restrict__ ad = agg + (size_t)d * D;
  for (int c = threadIdx.x; c < D; c += blockDim.x)
    atomicAdd(&ad[c], hs[c]);
}

// -------------------------------------------------- fused SAGE layer (WMMA)
// out = relu( l2norm_row( (agg/max(deg,1)) @ Wl + b + hin @ Wr ) )
// block = 128 threads (4 waves), computes 16 rows x 256 cols
template <int K>
__global__ void __launch_bounds__(128)
sage_layer(const float* __restrict__ agg, const float* __restrict__ hin,
           const float* __restrict__ deg, const float* __restrict__ Wl,
           const float* __restrict__ Wr, const float* __restrict__ bias,
           float* __restrict__ hout) {
  constexpr int P = K + 4;  // padded LDS stride -> conflict-free column reads
  __shared__ float sA[16 * P];
  __shared__ float sX[16 * P];
  __shared__ float invd[16];
  __shared__ float rowsum[16];

  const int tid  = threadIdx.x;
  const int wave = tid >> 5;
  const int lane = tid & 31;
  const int half = lane >> 4;   // 0: lanes 0-15, 1: lanes 16-31
  const int lid  = lane & 15;
  const int row0 = blockIdx.x * 16;

  if (tid < 16) {
    invd[tid]   = 1.0f / fmaxf(deg[row0 + tid], 1.0f);
    rowsum[tid] = 0.0f;
  }
  __syncthreads();

  // stage 16xK strips of (agg * invdeg) and hin into LDS
  for (int idx = tid; idx < 16 * K; idx += 128) {
    int r = idx / K;
    int c = idx - r * K;
    sA[r * P + c] = agg[(size_t)(row0 + r) * K + c] * invd[r];
    sX[r * P + c] = hin[(size_t)(row0 + r) * K + c];
  }
  __syncthreads();

  v8f acc[4] = {};               // 4 column tiles of 16 per wave -> 64 cols/wave
  const int nb = wave * 64 + lid;  // this lane's column within each tile base

  for (int k0 = 0; k0 < K; k0 += 4) {
    const int kk = k0 + 2 * half;  // ISA A/B fragment layout: VGPR0={K|K+... }
    v2f aA = { sA[lid * P + kk], sA[lid * P + kk + 1] };
    v2f aX = { sX[lid * P + kk], sX[lid * P + kk + 1] };
    const float* __restrict__ wl = Wl + (size_t)kk * HID + nb;
    const float* __restrict__ wr = Wr + (size_t)kk * HID + nb;
#pragma unroll
    for (int t = 0; t < 4; ++t) {
      v2f bL = { wl[16 * t], wl[HID + 16 * t] };
      v2f bR = { wr[16 * t], wr[HID + 16 * t] };
      acc[t] = WMMA_F32(aA, bL, acc[t]);
      acc[t] = WMMA_F32(aX, bR, acc[t]);
    }
  }

  // + bias
#pragma unroll
  for (int t = 0; t < 4; ++t) {
    float bv = bias[nb + 16 * t];
#pragma unroll
    for (int i = 0; i < 8; ++i) acc[t][i] += bv;
  }

  // row-wise sum of squares (C/D layout: lane holds rows i+8*half, col nb+16t)
#pragma unroll
  for (int i = 0; i < 8; ++i) {
    float p = 0.0f;
#pragma unroll
    for (int t = 0; t < 4; ++t) p += acc[t][i] * acc[t][i];
    atomicAdd(&rowsum[i + 8 * half], p);
  }
  __syncthreads();

  // scale = 1/max(||row||, eps), relu, store
#pragma unroll
  for (int i = 0; i < 8; ++i) {
    int r = i + 8 * half;
    float inv = 1.0f / fmaxf(sqrtf(rowsum[r]), 1e-12f);
    size_t base = (size_t)(row0 + r) * HID;
#pragma unroll
    for (int t = 0; t < 4; ++t)
      hout[base + nb + 16 * t] = fmaxf(acc[t][i] * inv, 0.0f);
  }
}

// ------------------------------------------------------------- max pooling
// post-relu values are >= 0, so unsigned-bit atomicMax == float max, and the
// 0-initialized buffer reproduces the "empty graph -> 0" isfinite fixup.
__global__ void pool_max(const float* __restrict__ h, const int* __restrict__ batch,
                         unsigned int* __restrict__ pooled) {
  size_t idx = (size_t)blockIdx.x * blockDim.x + threadIdx.x;
  int n = (int)(idx >> 8);
  int c = (int)(idx & 255);
  if (n < N_NODES) {
    int g = batch[n];
    float v = h[(size_t)n * HID + c];
    atomicMax(&pooled[(size_t)g * HID + c], __float_as_uint(v));
  }
}

// ------------------------------------------------------- output projection
__global__ void out_proj(const float* __restrict__ pooled, const float* __restrict__ Wout,
                         const float* __restrict__ bout, float* __restrict__ out) {
  int idx = blockIdx.x * blockDim.x + threadIdx.x;  // g*OUT_CH + o
  if (idx >= N_GRAPHS * OUT_CH) return;
  int g = idx / OUT_CH;
  int o = idx - g * OUT_CH;
  float s = bout[o];
  const float* __restrict__ pr = pooled + (size_t)g * HID;
  for (int c = 0; c < HID; ++c) s += pr[c] * Wout[c * OUT_CH + o];
  out[idx] = s;
}

// ------------------------------------------------------------------ driver
extern "C" void kernel_launch(void* const* d_in, const int* in_sizes, int n_in,
                              void* d_out, int out_size, void* d_ws, size_t ws_size,
                              hipStream_t stream) {
  (void)in_sizes; (void)n_in; (void)out_size; (void)ws_size;

  const float* x     = (const float*)d_in[0];
  const int*   ei    = (const int*)d_in[1];
  const int*   src   = ei;
  const int*   dst   = ei + N_EDGES;
  const int*   batch = (const int*)d_in[2];
  const float* Wl0 = (const float*)d_in[3];
  const float* Wr0 = (const float*)d_in[4];
  const float* b0  = (const float*)d_in[5];
  const float* Wl1 = (const float*)d_in[6];
  const float* Wr1 = (const float*)d_in[7];
  const float* b1  = (const float*)d_in[8];
  const float* Wl2 = (const float*)d_in[9];
  const float* Wr2 = (const float*)d_in[10];
  const float* b2  = (const float*)d_in[11];
  const float* Wout = (const float*)d_in[12];
  const float* bout = (const float*)d_in[13];

  // workspace carve-up (floats)
  float* ws  = (float*)d_ws;
  float* deg = ws;                                   // N
  float* agg = deg + N_NODES;                        // N*256
  float* hA  = agg + (size_t)N_NODES * HID;          // N*256
  float* hB  = hA + (size_t)N_NODES * HID;           // N*256
  unsigned int* pooled = (unsigned int*)(hB + (size_t)N_NODES * HID);  // G*256

  const int ROWTILES = N_NODES / 16;  // 3125 (exact)

  auto zero = [&](float* p, size_t n) {
    zero_f<<<(unsigned)((n + 255) / 256), 256, 0, stream>>>(p, n);
  };

  // degrees (same edge set for all layers)
  zero(deg, N_NODES);
  compute_deg<<<(N_EDGES + 255) / 256, 256, 0, stream>>>(dst, deg, N_EDGES);

  // layer 0: x (128) -> hA (256)
  zero(agg, (size_t)N_NODES * IN_CH);
  scatter_add<<<N_EDGES, 128, 0, stream>>>(x, src, dst, agg, IN_CH);
  sage_layer<IN_CH><<<ROWTILES, 128, 0, stream>>>(agg, x, deg, Wl0, Wr0, b0, hA);

  // layer 1: hA -> hB
  zero(agg, (size_t)N_NODES * HID);
  scatter_add<<<N_EDGES, 128, 0, stream>>>(hA, src, dst, agg, HID);
  sage_layer<HID><<<ROWTILES, 128, 0, stream>>>(agg, hA, deg, Wl1, Wr1, b1, hB);

  // layer 2: hB -> hA
  zero(agg, (size_t)N_NODES * HID);
  scatter_add<<<N_EDGES, 128, 0, stream>>>(hB, src, dst, agg, HID);
  sage_layer<HID><<<ROWTILES, 128, 0, stream>>>(agg, hB, deg, Wl2, Wr2, b2, hA);

  // global max pool + output projection
  zero((float*)pooled, (size_t)N_GRAPHS * HID);
  {
    size_t nthreads = (size_t)N_NODES * HID;
    pool_max<<<(unsigned)((nthreads + 255) / 256), 256, 0, stream>>>(hA, batch, pooled);
  }
  out_proj<<<(N_GRAPHS * OUT_CH + 127) / 128, 128, 0, stream>>>(
      (const float*)pooled, Wout, bout, (float*)d_out);
}